// PagedAttention_12309376270526
// MI455X (gfx1250) — compile-verified
//
#include <hip/hip_runtime.h>
#include <hip/hip_bf16.h>
#include <stdint.h>

// ---------------- problem constants (match reference) ----------------
#define T_SEQS 32
#define HIDDEN 4096
#define NHEADS 32
#define KVH 8
#define GROUP 4          // NHEADS / KVH
#define HD 128           // head dim
#define BS 16            // paged block size
#define MAX_BLOCKS 128
#define QKV_N (NHEADS*HD + 2*KVH*HD)   // 6144
#define KTILE 64

typedef __attribute__((ext_vector_type(2))) float v2f;
typedef __attribute__((ext_vector_type(8))) float v8f;
typedef __attribute__((ext_vector_type(4))) unsigned int v4u;
typedef __attribute__((ext_vector_type(8))) int v8i_t;
typedef __attribute__((ext_vector_type(4))) int v4i_t;

// ---------------- fp32 WMMA: D = A(16x4) * B(4x16) + C(16x16) ----------------
__device__ __forceinline__ v8f wmma4(v2f a, v2f b, v8f c) {
  return __builtin_amdgcn_wmma_f32_16x16x4_f32(
      /*neg_a=*/false, a, /*neg_b=*/false, b,
      /*c_mod=*/(short)0, c, /*reuse_a=*/false, /*reuse_b=*/false);
}

// ---------------- TDM: 2D tile (fp32) global -> LDS ----------------
// Builds D# group0/group1 per CDNA5 ISA bitfields (count=1, type=2,
// data_size=4B, tile = tile_d0 x tile_d1, row stride d0_stride elems).
__device__ __forceinline__ void tdm_load_2d(unsigned lds_byte_off,
                                            const void* gptr,
                                            unsigned tile_d0, unsigned tile_d1,
                                            unsigned tensor_d0, unsigned tensor_d1,
                                            unsigned d0_stride_elems) {
  unsigned long long ga = (unsigned long long)gptr;
  v4u g0;
  g0[0] = 1u;                                            // count=1 (valid)
  g0[1] = lds_byte_off;                                  // lds_addr
  g0[2] = (unsigned)ga;                                  // global_addr[31:0]
  g0[3] = (unsigned)((ga >> 32) & 0x01FFFFFFu) | (2u << 30); // [56:32] | type=2
  v8i_t g1;
  g1[0] = (int)(2u << 16);                               // data_size=2 -> 4B; wg_mask=0
  g1[1] = (int)((tensor_d0 & 0xFFFFu) << 16);            // tensor_dim0[15:0]
  g1[2] = (int)((tensor_d0 >> 16) | ((tensor_d1 & 0xFFFFu) << 16));
  g1[3] = (int)((tensor_d1 >> 16) | ((tile_d0 & 0xFFFFu) << 16));
  g1[4] = (int)(tile_d1 & 0xFFFFu);                      // tile_dim2 = 0
  g1[5] = (int)d0_stride_elems;                          // dim0_stride[31:0]
  g1[6] = 0;                                             // dim0_stride[47:32], dim1_stride lo
  g1[7] = 0;                                             // dim1_stride hi
  v4i_t g2 = {0, 0, 0, 0};
  v4i_t g3 = {0, 0, 0, 0};
  v8i_t gx = {0, 0, 0, 0, 0, 0, 0, 0};
  __builtin_amdgcn_tensor_load_to_lds(g0, g1, g2, g3, gx, 0);
}

// ---------------- GEMM: C[32 x N] = A[32 x K] * B[K x N] ----------------
// grid = (N/16); one wave computes a full 32x16 C tile so the weight stream
// (the dominant HBM traffic) is read exactly once. Both A (32 x KTILE) and
// B (KTILE x 16) slabs are staged into LDS by the TDM, double-buffered.
__global__ __launch_bounds__(32) void gemm32_wmma(const float* __restrict__ A,
                                                  const float* __restrict__ B,
                                                  float* __restrict__ C,
                                                  int N, int K, int ldc) {
  const int n0 = blockIdx.x * 16;
  const int lane = threadIdx.x;
  const int half = lane >> 4;
  const int l16 = lane & 15;

  __shared__ float at[2][32 * KTILE];   // A slabs: 32 rows x KTILE
  __shared__ float bt[2][KTILE * 16];   // B slabs: KTILE rows x 16 cols

  v8f acc0 = {0.f, 0.f, 0.f, 0.f, 0.f, 0.f, 0.f, 0.f};   // rows 0..15
  v8f acc1 = {0.f, 0.f, 0.f, 0.f, 0.f, 0.f, 0.f, 0.f};   // rows 16..31

  // prologue: stage slab 0 (A and B)
  tdm_load_2d((unsigned)(uintptr_t)&at[0][0], A,
              KTILE, 32, (unsigned)K, 32, (unsigned)K);
  tdm_load_2d((unsigned)(uintptr_t)&bt[0][0], B + n0,
              16, KTILE, 16, KTILE, (unsigned)N);

  for (int k0 = 0; k0 < K; k0 += KTILE) {
    const int buf = (k0 / KTILE) & 1;
    if (k0 + KTILE < K) {
      tdm_load_2d((unsigned)(uintptr_t)&at[buf ^ 1][0], A + k0 + KTILE,
                  KTILE, 32, (unsigned)K, 32, (unsigned)K);
      tdm_load_2d((unsigned)(uintptr_t)&bt[buf ^ 1][0],
                  B + (size_t)(k0 + KTILE) * N + n0,
                  16, KTILE, 16, KTILE, (unsigned)N);
      __builtin_amdgcn_s_wait_tensorcnt(2);   // the 2 loads of current slab done
    } else {
      __builtin_amdgcn_s_wait_tensorcnt(0);
    }
    __syncthreads();

#pragma unroll 8
    for (int kk = 0; kk < KTILE; kk += 4) {
      const int kg = kk + half * 2;    // K offset within slab for VGPR pair
      // B fragment: lane (N=l16), VGPR j holds K row kg + j
      v2f b;
      b[0] = bt[buf][(kg + 0) * 16 + l16];
      b[1] = bt[buf][(kg + 1) * 16 + l16];
      // A fragments for both M tiles (share B)
      float2 a2 = *(const float2*)&at[buf][l16 * KTILE + kg];
      v2f a0; a0[0] = a2.x; a0[1] = a2.y;
      float2 a3 = *(const float2*)&at[buf][(l16 + 16) * KTILE + kg];
      v2f a1; a1[0] = a3.x; a1[1] = a3.y;
      acc0 = wmma4(a0, b, acc0);
      acc1 = wmma4(a1, b, acc1);
    }
    __syncthreads();   // LDS reads done before TDM rewrites this buffer
  }

  // C layout: VGPR r -> (M = r + half*8, N = l16)
#pragma unroll
  for (int r = 0; r < 8; ++r) {
    C[(size_t)(r + half * 8) * ldc + n0 + l16] = acc0[r];
    C[(size_t)(16 + r + half * 8) * ldc + n0 + l16] = acc1[r];
  }
}

// ---------------- RoPE in-place on q,k of the qkv workspace ----------------
__global__ void rope_kernel(float* __restrict__ qkv, const int* __restrict__ ctx) {
  const int t = blockIdx.x;
  const float pos = (float)(ctx[t] - 1);
  // q heads 0..31 at head*128, k heads 32..39 land at 4096 + (h-32)*128
  for (int idx = threadIdx.x; idx < (NHEADS + KVH) * (HD / 2); idx += blockDim.x) {
    const int head = idx / (HD / 2);
    const int i = idx % (HD / 2);
    const float inv = __powf(10000.f, -(2.f * (float)i) / (float)HD);
    const float f = pos * inv;
    const float c = __cosf(f), s = __sinf(f);
    float* base = qkv + (size_t)t * QKV_N + head * HD;
    const float x1 = base[i], x2 = base[i + HD / 2];
    base[i]          = x1 * c - x2 * s;
    base[i + HD / 2] = x2 * c + x1 * s;
  }
}

// ---------------- paged attention, online softmax, 1 wave / (seq, kvh) ----------------
__global__ __launch_bounds__(32) void paged_attn_kernel(
    const float* __restrict__ qkv, const float* __restrict__ k_cache,
    const float* __restrict__ v_cache, const int* __restrict__ block_tables,
    const int* __restrict__ ctx_lens, float* __restrict__ attn_out) {
  const int t = blockIdx.x;     // sequence
  const int h = blockIdx.y;     // kv head
  const int lane = threadIdx.x;
  const int half = lane >> 4;
  const int l16 = lane & 15;
  const int ctx = ctx_lens[t];

  __shared__ float qs[GROUP * HD];     // 4 query heads x 128
  __shared__ float kb[BS * HD];        // K tile (16 tokens x 128)
  __shared__ float vb[BS * HD];        // V tile
  __shared__ float ps[16 * 16];        // P transposed to A-layout

  const float* qbase = qkv + (size_t)t * QKV_N + (h * GROUP) * HD;
  const float* knew  = qkv + (size_t)t * QKV_N + NHEADS * HD + h * HD;
  const float* vnew  = qkv + (size_t)t * QKV_N + NHEADS * HD + KVH * HD + h * HD;
  for (int i = lane; i < GROUP * HD; i += 32) qs[i] = qbase[i];
  __syncthreads();

  float m_r[8], l_r[8];
  v8f o[8];
#pragma unroll
  for (int r = 0; r < 8; ++r) { m_r[r] = -3.0e38f; l_r[r] = 0.f; }
#pragma unroll
  for (int nt = 0; nt < 8; ++nt) {
    v8f z = {0.f, 0.f, 0.f, 0.f, 0.f, 0.f, 0.f, 0.f};
    o[nt] = z;
  }

  const int ntiles = (ctx + BS - 1) / BS;
  const int last = ctx - 1, jlast = last / BS, slot = last % BS;
  const float scale = 0.08838834764831845f;  // 1/sqrt(128)

  for (int j = 0; j < ntiles; ++j) {
    __syncthreads();   // previous-tile LDS reads complete before TDM overwrite
    const int phys = block_tables[t * MAX_BLOCKS + j];
    // one paged block == one 16-token WMMA tile: 2D tile, rows strided KVH*HD
    const float* kg = k_cache + ((size_t)phys * BS * KVH + h) * HD;
    const float* vg = v_cache + ((size_t)phys * BS * KVH + h) * HD;
    tdm_load_2d((unsigned)(uintptr_t)kb, kg, HD, BS, HD, BS, KVH * HD);
    tdm_load_2d((unsigned)(uintptr_t)vb, vg, HD, BS, HD, BS, KVH * HD);
    __builtin_amdgcn_s_wait_tensorcnt(0);
    __syncthreads();

    if (j == jlast) {  // splice the fresh (RoPE'd) K/V of the new token (no cache mutation)
      for (int i = lane; i < HD; i += 32) {
        kb[slot * HD + i] = knew[i];
        vb[slot * HD + i] = vnew[i];
      }
      __syncthreads();
    }

    // ---- scores S[16x16] = Q(A: M=head row, K=dim) * K^T(B: K=dim, N=token)
    v8f s = {0.f, 0.f, 0.f, 0.f, 0.f, 0.f, 0.f, 0.f};
#pragma unroll 8
    for (int kk = 0; kk < HD; kk += 4) {
      v2f a;
      if (l16 < GROUP) {
        float2 a2 = *(const float2*)&qs[l16 * HD + kk + half * 2];
        a[0] = a2.x; a[1] = a2.y;
      } else { a[0] = 0.f; a[1] = 0.f; }
      float2 b2 = *(const float2*)&kb[l16 * HD + kk + half * 2];
      v2f b; b[0] = b2.x; b[1] = b2.y;
      s = wmma4(a, b, s);
    }

    // ---- online softmax (row max/sum across 16 lanes via wave32 butterflies)
    const int tok = j * BS + l16;
    const bool valid = tok < ctx;
    float pr[8], corr[8];
#pragma unroll
    for (int r = 0; r < 8; ++r) {
      float sv = valid ? s[r] * scale : -3.0e38f;
      float tmax = sv;
      tmax = fmaxf(tmax, __shfl_xor(tmax, 8, 32));
      tmax = fmaxf(tmax, __shfl_xor(tmax, 4, 32));
      tmax = fmaxf(tmax, __shfl_xor(tmax, 2, 32));
      tmax = fmaxf(tmax, __shfl_xor(tmax, 1, 32));
      const float mnew = fmaxf(m_r[r], tmax);
      const float c = __expf(m_r[r] - mnew);
      float p = __expf(sv - mnew);
      if (!valid) p = 0.f;
      float psum = p;
      psum += __shfl_xor(psum, 8, 32);
      psum += __shfl_xor(psum, 4, 32);
      psum += __shfl_xor(psum, 2, 32);
      psum += __shfl_xor(psum, 1, 32);
      l_r[r] = l_r[r] * c + psum;
      m_r[r] = mnew;
      corr[r] = c;
      pr[r] = p;
    }

    // transpose P (C-layout -> row-major) through LDS for the A operand
    __syncthreads();
#pragma unroll
    for (int r = 0; r < 8; ++r)
      ps[(r + half * 8) * 16 + l16] = pr[r];
    __syncthreads();

    // rescale output accumulators
#pragma unroll
    for (int nt = 0; nt < 8; ++nt)
#pragma unroll
      for (int r = 0; r < 8; ++r) o[nt][r] *= corr[r];

    // ---- O += P(16x16) * V(16x128): 8 N-tiles, K=16 in chunks of 4
#pragma unroll
    for (int nt = 0; nt < 8; ++nt) {
#pragma unroll
      for (int kk = 0; kk < 16; kk += 4) {
        float2 a2 = *(const float2*)&ps[l16 * 16 + kk + half * 2];
        v2f a; a[0] = a2.x; a[1] = a2.y;
        v2f b;
        b[0] = vb[(kk + half * 2 + 0) * HD + nt * 16 + l16];
        b[1] = vb[(kk + half * 2 + 1) * HD + nt * 16 + l16];
        o[nt] = wmma4(a, b, o[nt]);
      }
    }
  }

  // ---- epilogue: rows 0..3 (valid heads) live in lanes 0..15, VGPRs 0..3
  if (half == 0) {
#pragma unroll
    for (int r = 0; r < GROUP; ++r) {
      const float inv_l = 1.f / l_r[r];
      const int head = h * GROUP + r;
#pragma unroll
      for (int nt = 0; nt < 8; ++nt)
        attn_out[(size_t)t * (NHEADS * HD) + head * HD + nt * 16 + l16] =
            o[nt][r] * inv_l;
    }
  }
}

// ---------------- host-side launch ----------------
extern "C" void kernel_launch(void* const* d_in, const int* in_sizes, int n_in,
                              void* d_out, int out_size, void* d_ws, size_t ws_size,
                              hipStream_t stream) {
  const float* hidden = (const float*)d_in[0];
  const float* Wq = (const float*)d_in[1];
  const float* Wk = (const float*)d_in[2];
  const float* Wv = (const float*)d_in[3];
  const float* Wo = (const float*)d_in[4];
  const float* k_cache = (const float*)d_in[5];
  const float* v_cache = (const float*)d_in[6];
  const int* block_tables = (const int*)d_in[7];
  const int* ctx_lens = (const int*)d_in[8];
  float* out = (float*)d_out;

  float* qkv  = (float*)d_ws;                       // [32, 6144]
  float* attn = qkv + (size_t)T_SEQS * QKV_N;       // [32, 4096]

  // QKV projections (fp32 WMMA, TDM-staged A and B slabs, weights read once)
  gemm32_wmma<<<NHEADS * HD / 16, 32, 0, stream>>>(
      hidden, Wq, qkv, NHEADS * HD, HIDDEN, QKV_N);
  gemm32_wmma<<<KVH * HD / 16, 32, 0, stream>>>(
      hidden, Wk, qkv + NHEADS * HD, KVH * HD, HIDDEN, QKV_N);
  gemm32_wmma<<<KVH * HD / 16, 32, 0, stream>>>(
      hidden, Wv, qkv + NHEADS * HD + KVH * HD, KVH * HD, HIDDEN, QKV_N);

  // RoPE on q and k (in workspace; caches are never mutated)
  rope_kernel<<<T_SEQS, 128, 0, stream>>>(qkv, ctx_lens);

  // Paged attention: one wave per (sequence, kv head)
  paged_attn_kernel<<<dim3(T_SEQS, KVH), 32, 0, stream>>>(
      qkv, k_cache, v_cache, block_tables, ctx_lens, attn);

  // Output projection
  gemm32_wmma<<<HIDDEN / 16, 32, 0, stream>>>(
      attn, Wo, out, HIDDEN, HIDDEN, HIDDEN);
}